// MyLoss_80255758893338
// MI455X (gfx1250) — compile-verified
//
#include <hip/hip_runtime.h>

// ---- vector types ----------------------------------------------------------
typedef float f4  __attribute__((ext_vector_type(4)));
typedef int   i4  __attribute__((ext_vector_type(4)));
typedef float v2f __attribute__((ext_vector_type(2)));
typedef float v8f __attribute__((ext_vector_type(8)));

// ---- wave32 reduction via V_WMMA_F32_16X16X4_F32 ---------------------------
// A (16x4 f32, 2 VGPRs): VGPR0 = per-lane partial -> (M=lane%16, K=0 for lanes
// 0-15, K=2 for lanes 16-31); VGPR1 = 0 (K=1,3). B = all-ones 4x16, so
// D[r][c] = p[r] + p[r+16] for every column c. Each lane sums its 8 D VGPRs
// (rows 0-7 or rows 8-15 of its column); one shfl_xor(16) completes the
// 32-lane total. All-f32: no precision loss vs the reference.
__device__ __forceinline__ float wave_reduce_wmma(float acc) {
    v2f a; a.x = acc;  a.y = 0.0f;
    v2f b; b.x = 1.0f; b.y = 1.0f;
    v8f c = {};
    c = __builtin_amdgcn_wmma_f32_16x16x4_f32(
            /*neg_a=*/false, a, /*neg_b=*/false, b,
            /*c_mod=*/(short)0, c, /*reuse_a=*/false, /*reuse_b=*/false);
    float s = ((c[0] + c[1]) + (c[2] + c[3])) + ((c[4] + c[5]) + (c[6] + c[7]));
    s += __shfl_xor(s, 16, 32);   // rows 0-7 + rows 8-15
    return s;                      // every lane holds the wave total
}

// Branchless per-element contribution:
//   label==1 -> -log(o)*w1 ; label==0 -> (1+log(o))*w0 ; else -> 0
// Predicates become float masks (v_cmp + v_cndmask w/ inline 1.0), folded with
// two FMAs -- no ternaries left for the compiler to turn into saveexec branches.
__device__ __forceinline__ float accum_elem(float acc, float o, int lb,
                                            float w0, float w1) {
    float l  = __logf(o);                    // v_log_f32 * ln(2)
    float v1 = -l * w1;                      // label == 1 value
    float v0 = (1.0f + l) * w0;              // label == 0 value
    float m1 = (lb == 1) ? 1.0f : 0.0f;      // v_cndmask, inline 1.0
    float m0 = (lb == 0) ? 1.0f : 0.0f;
    return __builtin_fmaf(m1, v1, __builtin_fmaf(m0, v0, acc));
}

// ---- kernel 1: streaming pass, one f32 partial per block -------------------
__global__ void __launch_bounds__(256)
loss_partial_kernel(const float* __restrict__ out,
                    const int*   __restrict__ label,
                    const float* __restrict__ w,
                    float*       __restrict__ partials,
                    int n4)
{
    const float w0 = w[0];                   // uniform -> s_load
    const float w1 = w[1];

    const f4* out4 = (const f4*)out;
    const i4* lab4 = (const i4*)label;

    const int tid = (int)(blockIdx.x * blockDim.x + threadIdx.x);
    const int S   = (int)(gridDim.x * blockDim.x);

    // 4-deep unroll: 8 global_load_b128 in flight per trip, 4 independent
    // accumulator chains. For B = 2^25 / (2048*256 threads): exactly 4 trips.
    float a0 = 0.0f, a1 = 0.0f, a2 = 0.0f, a3 = 0.0f;
    int i = tid;
    for (; i + 3 * S < n4; i += 4 * S) {
        f4 o0 = __builtin_nontemporal_load(out4 + i);
        f4 o1 = __builtin_nontemporal_load(out4 + i + S);
        f4 o2 = __builtin_nontemporal_load(out4 + i + 2 * S);
        f4 o3 = __builtin_nontemporal_load(out4 + i + 3 * S);
        i4 l0 = __builtin_nontemporal_load(lab4 + i);
        i4 l1 = __builtin_nontemporal_load(lab4 + i + S);
        i4 l2 = __builtin_nontemporal_load(lab4 + i + 2 * S);
        i4 l3 = __builtin_nontemporal_load(lab4 + i + 3 * S);
#pragma unroll
        for (int j = 0; j < 4; ++j) {
            a0 = accum_elem(a0, o0[j], l0[j], w0, w1);
            a1 = accum_elem(a1, o1[j], l1[j], w0, w1);
            a2 = accum_elem(a2, o2[j], l2[j], w0, w1);
            a3 = accum_elem(a3, o3[j], l3[j], w0, w1);
        }
    }
    for (; i < n4; i += S) {                 // vector remainder (rarely taken)
        f4 o = __builtin_nontemporal_load(out4 + i);
        i4 l = __builtin_nontemporal_load(lab4 + i);
#pragma unroll
        for (int j = 0; j < 4; ++j)
            a0 = accum_elem(a0, o[j], l[j], w0, w1);
    }
    float acc = (a0 + a1) + (a2 + a3);

    // EXEC reconverged to all-1s here (full waves) -> WMMA legal
    float s = wave_reduce_wmma(acc);

    __shared__ float smem[8];                // 256 threads = 8 waves
    const int lane = threadIdx.x & 31;
    const int wid  = threadIdx.x >> 5;
    if (lane == 0) smem[wid] = s;
    __syncthreads();
    if (threadIdx.x == 0) {
        float t = 0.0f;
#pragma unroll
        for (int j = 0; j < 8; ++j) t += smem[j];
        partials[blockIdx.x] = t;
    }
}

// ---- kernel 2: deterministic single-block finish (+ scalar tail) -----------
__global__ void __launch_bounds__(256)
reduce_partials_kernel(const float* __restrict__ partials, int count,
                       const float* __restrict__ out,
                       const int*   __restrict__ label,
                       const float* __restrict__ w,
                       int tail_start, int n,
                       float* __restrict__ dst)
{
    float acc = 0.0f;
    for (int i = (int)threadIdx.x; i < count; i += 256)
        acc += partials[i];

    // tail elements (none for B = 2^25; spread across the block if any)
    if (tail_start < n) {
        const float w0 = w[0], w1 = w[1];
        for (int i = tail_start + (int)threadIdx.x; i < n; i += 256)
            acc = accum_elem(acc, out[i], label[i], w0, w1);
    }

    float s = wave_reduce_wmma(acc);

    __shared__ float smem[8];
    const int lane = threadIdx.x & 31;
    const int wid  = threadIdx.x >> 5;
    if (lane == 0) smem[wid] = s;
    __syncthreads();
    if (threadIdx.x == 0) {
        float t = 0.0f;
#pragma unroll
        for (int j = 0; j < 8; ++j) t += smem[j];
        dst[0] = t;                          // plain store: deterministic
    }
}

// ---- host-side launcher ----------------------------------------------------
extern "C" void kernel_launch(void* const* d_in, const int* in_sizes, int n_in,
                              void* d_out, int out_size, void* d_ws, size_t ws_size,
                              hipStream_t stream)
{
    const float* out   = (const float*)d_in[0];
    const int*   label = (const int*)  d_in[1];
    const float* w     = (const float*)d_in[2];
    float*       dst   = (float*)d_out;

    const int n  = in_sizes[0];              // 2^25
    const int n4 = n >> 2;

    const int threads = 256;
    int blocks = 2048;                       // 0.5M threads, 64 elems/thread
    float* partials = (float*)d_ws;
    const size_t max_parts = ws_size / sizeof(float);
    if ((size_t)blocks > max_parts) blocks = (int)max_parts;
    if (blocks < 1) {                        // degenerate workspace: stage in d_out
        blocks = 1;
        partials = dst;
    }

    loss_partial_kernel<<<blocks, threads, 0, stream>>>(out, label, w, partials, n4);
    reduce_partials_kernel<<<1, threads, 0, stream>>>(partials, blocks,
                                                      out, label, w,
                                                      n4 << 2, n, dst);
}